// PoseOptimizerLayer_50989851738446
// MI455X (gfx1250) — compile-verified
//
#include <hip/hip_runtime.h>

typedef float v2f __attribute__((ext_vector_type(2)));
typedef float v8f __attribute__((ext_vector_type(8)));

#define NA 1024
#define NB 1024
#define NBATCH 8
#define JGROUPS 32          // 32 column-groups of 32 columns (one 128B line per row)
#define ROWS_PER_WAVE 128   // 8 waves * 128 = 1024 rows

// Kernel 1: per (batch, column-group) compute partial M[p][q] = sum_ij U_p(i) A[i,j] V_q(j)
// via V_WMMA_F32_16X16X4_F32:  A-op = U^T (16 feature rows x K=4 i's, rows>=4 zero),
//                              B-op = streamed association data (4 i-rows x 16 j-cols).
__global__ __launch_bounds__(256)
void pose_q_reduce(const float* __restrict__ assoc,
                   const float* __restrict__ ptA,
                   const float* __restrict__ ptB,
                   float* __restrict__ Mpart) {
  __shared__ __align__(16) float sU[4 * NA];   // feature-major U table (16 KB)
  __shared__ float sM[8 * 16];

  const int b    = blockIdx.x;
  const int jg   = blockIdx.y;
  const int tid  = threadIdx.x;
  const int lane = tid & 31;
  const int wave = tid >> 5;
  const int n16  = lane & 15;     // WMMA column / feature-row index
  const int kh   = lane >> 4;     // K half: lanes 0-15 -> K=0,1 ; lanes 16-31 -> K=2,3

  // Phase 1: build U feature table in LDS
  const float* pa = ptA + (size_t)b * NA * 2;
  for (int r = tid; r < NA; r += 256) {
    float xa = pa[2 * r + 0], ya = pa[2 * r + 1];
    sU[0 * NA + r] = 1.0f;
    sU[1 * NA + r] = xa;
    sU[2 * NA + r] = ya;
    sU[3 * NA + r] = xa * xa + ya * ya;
  }
  __syncthreads();

  // This lane owns adjacent columns j0 (tile C0) and j0+1 (tile C1)
  const int j0 = jg * 32 + 2 * n16;
  const float4 pb = *(const float4*)(ptB + (size_t)b * NB * 2 + 2 * j0);
  const float V0[4] = {1.0f, pb.x, pb.y, pb.x * pb.x + pb.y * pb.y};
  const float V1[4] = {1.0f, pb.z, pb.w, pb.z * pb.z + pb.w * pb.w};

  const float* Ab = assoc + (size_t)b * NA * NB + j0;
  const int i_begin = wave * ROWS_PER_WAVE;

  const float umask = (n16 < 4) ? 1.0f : 0.0f;  // zero feature rows 4..15
  const int   prow  = n16 & 3;

  v8f C0 = {};
  v8f C1 = {};
  #pragma unroll 4
  for (int i = i_begin; i < i_begin + ROWS_PER_WAVE; i += 4) {
    const int r0 = i + 2 * kh;                       // rows r0, r0+1 for this lane's K pair
    // A-operand: U_p(r0), U_p(r0+1)  (one ds_load_b64, broadcast across column lanes)
    const v2f u = *(const v2f*)&sU[prow * NA + r0];
    v2f av; av[0] = u[0] * umask; av[1] = u[1] * umask;
    // B-operand: two full 128B-line coalesced b64 loads covering both column tiles
    const v2f d0 = *(const v2f*)(Ab + (size_t)r0 * NB);        // row r0  : cols j0, j0+1
    const v2f d1 = *(const v2f*)(Ab + (size_t)(r0 + 1) * NB);  // row r0+1: cols j0, j0+1
    v2f b0; b0[0] = d0[0]; b0[1] = d1[0];
    v2f b1; b1[0] = d0[1]; b1[1] = d1[1];
    C0 = __builtin_amdgcn_wmma_f32_16x16x4_f32(false, av, false, b0, (short)0, C0, false, false);
    C1 = __builtin_amdgcn_wmma_f32_16x16x4_f32(false, av, false, b1, (short)0, C1, false, false);
  }

  // Fold T[p, j_lane] with this lane's V features -> 16 accumulators (lanes 16-31 hold zeros)
  float acc[16];
  #pragma unroll
  for (int p = 0; p < 4; ++p)
    #pragma unroll
    for (int q = 0; q < 4; ++q)
      acc[p * 4 + q] = C0[p] * V0[q] + C1[p] * V1[q];

  // Deterministic reduction: wave butterfly, then cross-wave via LDS (no atomics)
  #pragma unroll
  for (int t = 0; t < 16; ++t) {
    float v = acc[t];
    #pragma unroll
    for (int m = 16; m >= 1; m >>= 1) v += __shfl_xor(v, m, 32);
    if (lane == 0) sM[wave * 16 + t] = v;
  }
  __syncthreads();
  if (tid < 16) {
    float s = 0.f;
    #pragma unroll
    for (int w = 0; w < 8; ++w) s += sM[w * 16 + tid];
    Mpart[((size_t)jg * NBATCH + b) * 16 + tid] = s;
  }
}

// Kernel 2: reduce the 32 column-group partials and assemble Q (8 x 5 x 5)
__global__ __launch_bounds__(32)
void pose_q_assemble(const float* __restrict__ Mpart, float* __restrict__ Q) {
  __shared__ float M[16];
  const int b = blockIdx.x;
  const int t = threadIdx.x;
  if (t < 16) {
    float s = 0.f;
    #pragma unroll
    for (int g = 0; g < JGROUPS; ++g) s += Mpart[((size_t)g * NBATCH + b) * 16 + t];
    M[t] = s;
  }
  __syncthreads();
  if (t < 25) {
    // Q[m][n] = c1*M[i1] + c2*M[i2], M indexed [p*4+q], p over {1,xa,ya,ra2}, q over {1,xb,yb,rb2}
    static const signed char i1[25] = {3, 1, 2, 5, 9,   1, 0, 0, 4, 8,
                                       2, 0, 0, 8, 4,   5, 4, 8, 12, 0,
                                       9, 8, 4, 0, 12};
    static const signed char i2[25] = {0, 0, 0, 10, 6,  0, 0, 0, 0, 0,
                                       0, 0, 0, 0, 0,   10, 0, 0, 0, 0,
                                       6, 0, 0, 0, 0};
    static const float c1[25] = { 1,-1,-1,-1, 1,  -1, 1, 0, 1,-1,
                                 -1, 0, 1, 1, 1,  -1, 1, 1, 1, 0,
                                  1,-1, 1, 0, 1};
    static const float c2[25] = { 0, 0, 0,-1,-1,   0, 0, 0, 0, 0,
                                  0, 0, 0, 0, 0,  -1, 0, 0, 0, 0,
                                 -1, 0, 0, 0, 0};
    Q[(size_t)b * 25 + t] = c1[t] * M[i1[t]] + c2[t] * M[i2[t]];
  }
}

extern "C" void kernel_launch(void* const* d_in, const int* in_sizes, int n_in,
                              void* d_out, int out_size, void* d_ws, size_t ws_size,
                              hipStream_t stream) {
  const float* assoc = (const float*)d_in[0];  // (8,1024,1024) f32
  const float* ptA   = (const float*)d_in[1];  // (8,1024,2)    f32
  const float* ptB   = (const float*)d_in[2];  // (8,1024,2)    f32
  float* Mpart = (float*)d_ws;                 // JGROUPS*8*16 floats = 16 KB
  float* Q     = (float*)d_out;                // 8*25 f32

  dim3 grid(NBATCH, JGROUPS);
  pose_q_reduce<<<grid, 256, 0, stream>>>(assoc, ptA, ptB, Mpart);
  pose_q_assemble<<<NBATCH, 32, 0, stream>>>(Mpart, Q);
}